// GraphFilterUp_9569187135849
// MI455X (gfx1250) — compile-verified
//
#include <hip/hip_runtime.h>

// Problem constants (match reference)
#define N_NODES 4096
#define K_TAPS  4
#define F_IN    8
#define F_OUT   16
#define BATCH   2
#define KTOT    (K_TAPS * N_NODES)   // 16384  flattened reduce dim
#define MROWS   (BATCH * F_OUT)      // 32     output rows
#define NWAVES  8                    // K-chunk waves per workgroup
#define KCHUNK  (KTOT / NWAVES)      // 2048 reduce elements per wave
#define PFROWS  64                   // prefetch distance in reduce rows

typedef float v2f __attribute__((ext_vector_type(2)));
typedef float v8f __attribute__((ext_vector_type(8)));

// ---------------------------------------------------------------------------
// Prep: build A (weights folded into x) pre-packed in WMMA A-fragment order.
//   A[row, K] = w[f,k] * x[b, f%8, m],  row = b*16+f, K = k*4096+m
// Packed so that a lane's A-fragment is one contiguous float2:
//   A2[(K>>1)*64 + row*2 + (K&1)] = A[row, K]
// ---------------------------------------------------------------------------
__global__ __launch_bounds__(256)
void gf_prep(const float* __restrict__ x,
             const float* __restrict__ weights,
             float* __restrict__ A2) {
    int idx = blockIdx.x * blockDim.x + threadIdx.x;   // 0 .. 32*16384-1
    int K   = idx >> 5;
    int row = idx & 31;
    int k   = K >> 12;              // K / 4096
    int m   = K & (N_NODES - 1);
    int b   = row >> 4;
    int f   = row & 15;
    float val = weights[f * K_TAPS + k] * x[((b * F_IN) + (f & 7)) * N_NODES + m];
    A2[((K >> 1) << 6) + row * 2 + (K & 1)] = val;
}

// ---------------------------------------------------------------------------
// Main: y[32, 4096] = A[32, 16384] @ SpowFlat[16384, 4096]
// One workgroup per 16-column N-tile; 8 waves each own a K-chunk of 2048 and
// accumulate two 16x16 f32 tiles (rows 0-15 and 16-31) with
// v_wmma_f32_16x16x4_f32, then reduce partials through LDS.
// Software-pipelined: fragments for step s+1 load while step s's WMMAs run.
// ---------------------------------------------------------------------------
__device__ __forceinline__ void gf_load_frags(const float* __restrict__ Bl,
                                              const float* __restrict__ Al,
                                              int kk, v2f& b, v2f& a0, v2f& a1) {
    b.x = Bl[(size_t)kk * N_NODES];
    b.y = Bl[(size_t)kk * N_NODES + N_NODES];
    const float* ap = Al + (kk >> 1) * 64;
    a0 = *(const v2f*)(ap);        // rows 0..15 fragment
    a1 = *(const v2f*)(ap + 32);   // rows 16..31 fragment
}

__global__ __launch_bounds__(NWAVES * 32)
void gf_wmma(const float* __restrict__ Spow,   // [16384, 4096] flat fp32
             const float* __restrict__ A2,     // packed A fragments
             float* __restrict__ y) {          // [32, 4096]
    const int lane    = threadIdx.x & 31;
    const int wave    = threadIdx.x >> 5;      // K-chunk id 0..7
    const int lane_lo = lane & 15;
    const int lane_hi = lane >> 4;
    const int n0      = blockIdx.x * 16;       // N-tile base column

    const int Kw = wave * KCHUNK;

    // B fragment element (per lane, per step kk):
    //   b[v] = Spow[(Kw + kk + 2*lane_hi + v) * 4096 + n0 + lane_lo]
    const float* Bl = Spow + (size_t)(Kw + 2 * lane_hi) * N_NODES + n0 + lane_lo;
    // A fragment: pair (K even, K odd) is a contiguous float2 per lane.
    const float* Al = A2 + (size_t)(Kw >> 1) * 64 + lane_hi * 64 + lane_lo * 2;

    v8f acc0 = {};   // rows 0..15
    v8f acc1 = {};   // rows 16..31

    v2f b, a0, a1;
    gf_load_frags(Bl, Al, 0, b, a0, a1);

    #pragma unroll 4
    for (int kk = 0; kk < KCHUNK; kk += 4) {
        // Cover all 4 B rows of the step 16 iterations ahead (rows are 16 KB
        // apart -> distinct cachelines; lanes 16-31 cover the +2/+3 rows).
        // Speculative prefetch: safe past the end of the chunk (dropped).
        __builtin_prefetch(Bl + (size_t)(kk + PFROWS) * N_NODES, 0, 3);
        __builtin_prefetch(Bl + (size_t)(kk + PFROWS + 1) * N_NODES, 0, 3);

        // Load next step's fragments while this step's WMMAs execute.
        int nk = kk + 4 < KCHUNK ? kk + 4 : kk;   // uniform scalar select
        v2f bn, a0n, a1n;
        gf_load_frags(Bl, Al, nk, bn, a0n, a1n);

        acc0 = __builtin_amdgcn_wmma_f32_16x16x4_f32(
                   false, a0, false, b, (short)0, acc0, false, false);
        acc1 = __builtin_amdgcn_wmma_f32_16x16x4_f32(
                   false, a1, false, b, (short)0, acc1, false, false);

        b = bn; a0 = a0n; a1 = a1n;
    }

    // Reduce the 8 K-chunk partials through LDS.
    __shared__ float red[NWAVES][2][8][32];   // 16 KB
    #pragma unroll
    for (int v = 0; v < 8; ++v) {
        red[wave][0][v][lane] = acc0[v];
        red[wave][1][v][lane] = acc1[v];
    }
    __syncthreads();

    if (wave < 2) {
        const int t = wave;                   // which 16-row M tile
        #pragma unroll
        for (int v = 0; v < 8; ++v) {
            float s = 0.f;
            #pragma unroll
            for (int w = 0; w < NWAVES; ++w) s += red[w][t][v][lane];
            // C/D layout: VGPR v -> M = v + 8*lane_hi, N = lane_lo
            int row = t * 16 + v + lane_hi * 8;
            y[(size_t)row * N_NODES + n0 + lane_lo] = s;
        }
    }
}

// ---------------------------------------------------------------------------
extern "C" void kernel_launch(void* const* d_in, const int* in_sizes, int n_in,
                              void* d_out, int out_size, void* d_ws, size_t ws_size,
                              hipStream_t stream) {
    const float* x       = (const float*)d_in[0];  // [2, 8, 4096]
    const float* Spow    = (const float*)d_in[1];  // [4, 4096, 4096]
    const float* weights = (const float*)d_in[2];  // [64]
    float* y  = (float*)d_out;                     // [2, 16, 4096]
    float* A2 = (float*)d_ws;                      // needs 32*16384*4 = 2 MB

    gf_prep<<<(MROWS * KTOT) / 256, 256, 0, stream>>>(x, weights, A2);
    gf_wmma<<<N_NODES / 16, NWAVES * 32, 0, stream>>>(Spow, A2, y);
}